// GATNet_61864708932309
// MI455X (gfx1250) — compile-verified
//
#include <hip/hip_runtime.h>

#define NEG_SLOPE 0.2f
#define EPSV 1e-16f

typedef float v2f __attribute__((ext_vector_type(2)));
typedef float v8f __attribute__((ext_vector_type(8)));

// ---------------------------------------------------------------- utilities

__global__ void fill_f32(float* __restrict__ p, float v, long long n) {
    long long i = (long long)blockIdx.x * blockDim.x + threadIdx.x;
    if (i < n) p[i] = v;
}

__global__ void reduce_sum(const float* __restrict__ x, long long n,
                           float* __restrict__ out) {
    __shared__ float s[256];
    long long stride = (long long)gridDim.x * blockDim.x;
    float acc = 0.f;
    for (long long i = (long long)blockIdx.x * blockDim.x + threadIdx.x; i < n;
         i += stride)
        acc += x[i];
    s[threadIdx.x] = acc;
    __syncthreads();
    for (int o = 128; o > 0; o >>= 1) {
        if ((int)threadIdx.x < o) s[threadIdx.x] += s[threadIdx.x + o];
        __syncthreads();
    }
    if (threadIdx.x == 0) atomicAdd(out, s[0]);
}

// scal layout: [0]=ew_sum (pre-accumulated), [1]=mean_ew, [2..3]=coef1, [4]=coef2
__global__ void scalars_kernel(float* __restrict__ scal, long long E,
                               const float* __restrict__ We1,
                               const float* __restrict__ ae1,
                               const float* __restrict__ We2,
                               const float* __restrict__ ae2) {
    if (blockIdx.x == 0 && threadIdx.x == 0) {
        scal[1] = scal[0] / (float)E;
        for (int h = 0; h < 2; ++h) {
            float c = 0.f;
            for (int i = 0; i < 64; ++i) c += We1[h * 64 + i] * ae1[h * 64 + i];
            scal[2 + h] = c;
        }
        float c2 = 0.f;
        for (int i = 0; i < 40; ++i) c2 += We2[i] * ae2[i];
        scal[4] = c2;
    }
}

__global__ void pad_w2(const float* __restrict__ W2, float* __restrict__ w2p) {
    int i = blockIdx.x * blockDim.x + threadIdx.x;
    if (i < 128 * 48) {
        int k = i / 48, c = i % 48;
        w2p[i] = (c < 40) ? W2[k * 40 + c] : 0.f;
    }
}

__device__ inline void atomicMaxFloat(float* addr, float val) {
    unsigned int* ua = (unsigned int*)addr;
    unsigned int old = *ua;
    while (true) {
        float oldf = __uint_as_float(old);
        if (oldf >= val) break;
        unsigned int assumed = old;
        old = atomicCAS(ua, assumed, __float_as_uint(val));
        if (old == assumed) break;
    }
}

// ---------------------------------------------------------------- WMMA GEMM
// C[M,N] = A[M,K] * B[K,N], row-major, fp32 via V_WMMA_F32_16X16X4_F32.
// One wave per 16x16 output tile: grid = (ceil(M/16), N/16), block = 32.
__global__ void __launch_bounds__(32) gemm_wmma_f32(
    const float* __restrict__ A, const float* __restrict__ B,
    float* __restrict__ C, int M, int N, int K) {
    int mt = blockIdx.x, nt = blockIdx.y;
    int lane = threadIdx.x;          // 0..31
    int half = lane >> 4;            // 0: K pair {0,1}, 1: K pair {2,3}
    int l = lane & 15;
    int row = mt * 16 + l;           // A row for this lane
    int arow = row < M ? row : (M - 1);
    int col = nt * 16 + l;           // B/C column for this lane
    const float* ap = A + (size_t)arow * K;

    v8f acc{};
#pragma unroll 4
    for (int k0 = 0; k0 < K; k0 += 4) {
        int ka = k0 + 2 * half;
        v2f a, b;
        a.x = ap[ka];
        a.y = ap[ka + 1];
        b.x = B[(size_t)ka * N + col];
        b.y = B[(size_t)(ka + 1) * N + col];
        acc = __builtin_amdgcn_wmma_f32_16x16x4_f32(
            false, a, false, b, (short)0, acc, false, false);
    }
#pragma unroll
    for (int v = 0; v < 8; ++v) {
        int r = mt * 16 + v + 8 * half;   // C/D layout: VGPR v -> row v (+8)
        if (r < M) C[(size_t)r * N + col] = acc[v];
    }
}

// ---------------------------------------------------------------- attention

__global__ void att_scores1(const float* __restrict__ h1,
                            const float* __restrict__ as,
                            const float* __restrict__ ad,
                            float* __restrict__ aS, float* __restrict__ aD,
                            long long N) {
    long long i = (long long)blockIdx.x * blockDim.x + threadIdx.x;  // N*2
    if (i >= N * 2) return;
    long long n = i >> 1;
    int h = (int)(i & 1);
    const float* hp = h1 + n * 128 + h * 64;
    float ss = 0.f, dd = 0.f;
    for (int c = 0; c < 64; ++c) {
        float v = hp[c];
        ss += v * as[h * 64 + c];
        dd += v * ad[h * 64 + c];
    }
    aS[i] = ss;
    aD[i] = dd;
}

__global__ void att_scores2(const float* __restrict__ h2,
                            const float* __restrict__ as,
                            const float* __restrict__ ad,
                            float* __restrict__ aS, float* __restrict__ aD,
                            long long N) {
    long long n = (long long)blockIdx.x * blockDim.x + threadIdx.x;
    if (n >= N) return;
    const float* hp = h2 + n * 48;  // padded stride 48, 40 valid
    float ss = 0.f, dd = 0.f;
    for (int c = 0; c < 40; ++c) {
        float v = hp[c];
        ss += v * as[c];
        dd += v * ad[c];
    }
    aS[n] = ss;
    aD[n] = dd;
}

// pass 1: alpha = leaky(a_s[src]+a_d[dst]+ew*coef); segment max into m
__global__ void edge_alpha(const int* __restrict__ src,
                           const int* __restrict__ dst,
                           const float* __restrict__ ew,
                           const float* __restrict__ meanw,
                           const float* __restrict__ coef,
                           const float* __restrict__ aS,
                           const float* __restrict__ aD,
                           float* __restrict__ alpha, float* __restrict__ m,
                           long long E, long long Etot, int H) {
    long long e = (long long)blockIdx.x * blockDim.x + threadIdx.x;
    if (e >= Etot) return;
    int s, d;
    float w;
    if (e < E) { s = src[e]; d = dst[e]; w = ew[e]; }
    else       { s = d = (int)(e - E);   w = meanw[0]; }
    for (int h = 0; h < H; ++h) {
        float al = aS[(size_t)s * H + h] + aD[(size_t)d * H + h] + w * coef[h];
        al = (al > 0.f) ? al : NEG_SLOPE * al;
        alpha[e * H + h] = al;
        atomicMaxFloat(&m[(size_t)d * H + h], al);
    }
}

// pass 2: alpha <- exp(alpha - m[dst]); segment sum into denom
__global__ void edge_exp(const int* __restrict__ dst,
                         float* __restrict__ alpha, const float* __restrict__ m,
                         float* __restrict__ denom, long long E, long long Etot,
                         int H) {
    long long e = (long long)blockIdx.x * blockDim.x + threadIdx.x;
    if (e >= Etot) return;
    int d = (e < E) ? dst[e] : (int)(e - E);
    for (int h = 0; h < H; ++h) {
        float ex = expf(alpha[e * H + h] - m[(size_t)d * H + h]);
        alpha[e * H + h] = ex;
        atomicAdd(&denom[(size_t)d * H + h], ex);
    }
}

// pass 3 (layer 1): agg[dst,h,c] += att * h1[src,h,c]; 128 threads per edge
__global__ void __launch_bounds__(128) edge_agg1(
    const int* __restrict__ src, const int* __restrict__ dst,
    const float* __restrict__ alpha, const float* __restrict__ denom,
    const float* __restrict__ h1, float* __restrict__ agg, long long E) {
    long long e = blockIdx.x;
    int t = threadIdx.x;  // channel 0..127, head = t>>6
    int s, d;
    if (e < E) { s = src[e]; d = dst[e]; }
    else       { s = d = (int)(e - E); }
    int h = t >> 6;
    float att = alpha[e * 2 + h] / (denom[(size_t)d * 2 + h] + EPSV);
    atomicAdd(&agg[(size_t)d * 128 + t], att * h1[(size_t)s * 128 + t]);
}

// pass 3 (layer 2): out2[dst,c] += att * h2[src,c]; 64 threads, 40 active
__global__ void __launch_bounds__(64) edge_agg2(
    const int* __restrict__ src, const int* __restrict__ dst,
    const float* __restrict__ alpha, const float* __restrict__ denom,
    const float* __restrict__ h2, float* __restrict__ out2, long long E) {
    long long e = blockIdx.x;
    int t = threadIdx.x;
    int s, d;
    if (e < E) { s = src[e]; d = dst[e]; }
    else       { s = d = (int)(e - E); }
    if (t < 40) {
        float att = alpha[e] / (denom[d] + EPSV);
        atomicAdd(&out2[(size_t)d * 40 + t], att * h2[(size_t)s * 48 + t]);
    }
}

__global__ void relu_bias(float* __restrict__ agg, const float* __restrict__ b1,
                          long long n) {
    long long i = (long long)blockIdx.x * blockDim.x + threadIdx.x;
    if (i < n) {
        float v = agg[i] + b1[i & 127];
        agg[i] = v > 0.f ? v : 0.f;
    }
}

__global__ void finalize(const float* __restrict__ out2,
                         const float* __restrict__ b2, float* __restrict__ out,
                         long long N) {
    long long n = (long long)blockIdx.x * blockDim.x + threadIdx.x;
    if (n >= N) return;
    const float* p = out2 + n * 40;
    float mx = -3.0e38f;
    for (int c = 0; c < 40; ++c) {
        float v = p[c] + b2[c];
        mx = v > mx ? v : mx;
    }
    float se = 0.f;
    for (int c = 0; c < 40; ++c) se += expf(p[c] + b2[c] - mx);
    float ls = logf(se);
    for (int c = 0; c < 40; ++c) out[n * 40 + c] = p[c] + b2[c] - mx - ls;
}

// ---------------------------------------------------------------- launch

extern "C" void kernel_launch(void* const* d_in, const int* in_sizes, int n_in,
                              void* d_out, int out_size, void* d_ws,
                              size_t ws_size, hipStream_t stream) {
    const float* x   = (const float*)d_in[0];
    const int*   ei  = (const int*)d_in[1];
    const float* ew  = (const float*)d_in[2];
    const float* W1  = (const float*)d_in[3];
    const float* as1 = (const float*)d_in[4];
    const float* ad1 = (const float*)d_in[5];
    const float* ae1 = (const float*)d_in[6];
    const float* We1 = (const float*)d_in[7];
    const float* b1  = (const float*)d_in[8];
    const float* W2  = (const float*)d_in[9];
    const float* as2 = (const float*)d_in[10];
    const float* ad2 = (const float*)d_in[11];
    const float* ae2 = (const float*)d_in[12];
    const float* We2 = (const float*)d_in[13];
    const float* b2  = (const float*)d_in[14];

    const long long N = in_sizes[0] / 128;
    const long long E = in_sizes[2];
    const long long Etot = E + N;
    const int* srcI = ei;
    const int* dstI = ei + E;

    float* ws = (float*)d_ws;
    size_t off = 0;
    auto alloc = [&](size_t cnt) { float* p = ws + off; off += cnt; return p; };
    float* h1     = alloc((size_t)N * 128);
    float* agg1   = alloc((size_t)N * 128);
    float* h2     = alloc((size_t)N * 48);
    float* out2   = alloc((size_t)N * 40);
    float* aS1    = alloc((size_t)N * 2);
    float* aD1    = alloc((size_t)N * 2);
    float* aS2    = alloc((size_t)N);
    float* aD2    = alloc((size_t)N);
    float* m1     = alloc((size_t)N * 2);
    float* dn1    = alloc((size_t)N * 2);
    float* m2     = alloc((size_t)N);
    float* dn2    = alloc((size_t)N);
    float* alpha1 = alloc((size_t)Etot * 2);
    float* alpha2 = alloc((size_t)Etot);
    float* w2p    = alloc(128 * 48);
    float* scal   = alloc(8);

    const int TB = 256;
    auto blks = [&](long long n) { return (unsigned)((n + TB - 1) / TB); };

    // init accumulators
    fill_f32<<<1, 8, 0, stream>>>(scal, 0.f, 8);
    fill_f32<<<blks(N * 128), TB, 0, stream>>>(agg1, 0.f, N * 128);
    fill_f32<<<blks(N * 40), TB, 0, stream>>>(out2, 0.f, N * 40);
    fill_f32<<<blks(N * 2), TB, 0, stream>>>(dn1, 0.f, N * 2);
    fill_f32<<<blks(N), TB, 0, stream>>>(dn2, 0.f, N);
    fill_f32<<<blks(N * 2), TB, 0, stream>>>(m1, -3.0e38f, N * 2);
    fill_f32<<<blks(N), TB, 0, stream>>>(m2, -3.0e38f, N);

    // scalars: mean edge weight, per-head edge coefficients
    reduce_sum<<<1024, TB, 0, stream>>>(ew, E, scal);
    scalars_kernel<<<1, 32, 0, stream>>>(scal, E, We1, ae1, We2, ae2);
    pad_w2<<<(128 * 48 + TB - 1) / TB, TB, 0, stream>>>(W2, w2p);

    // ---- layer 1 ----
    dim3 g1((unsigned)((N + 15) / 16), 8);
    gemm_wmma_f32<<<g1, 32, 0, stream>>>(x, W1, h1, (int)N, 128, 128);
    att_scores1<<<blks(N * 2), TB, 0, stream>>>(h1, as1, ad1, aS1, aD1, N);
    edge_alpha<<<blks(Etot), TB, 0, stream>>>(srcI, dstI, ew, scal + 1, scal + 2,
                                              aS1, aD1, alpha1, m1, E, Etot, 2);
    edge_exp<<<blks(Etot), TB, 0, stream>>>(dstI, alpha1, m1, dn1, E, Etot, 2);
    edge_agg1<<<(unsigned)Etot, 128, 0, stream>>>(srcI, dstI, alpha1, dn1, h1,
                                                  agg1, E);
    relu_bias<<<blks(N * 128), TB, 0, stream>>>(agg1, b1, N * 128);

    // ---- layer 2 ----
    dim3 g2((unsigned)((N + 15) / 16), 3);
    gemm_wmma_f32<<<g2, 32, 0, stream>>>(agg1, w2p, h2, (int)N, 48, 128);
    att_scores2<<<blks(N), TB, 0, stream>>>(h2, as2, ad2, aS2, aD2, N);
    edge_alpha<<<blks(Etot), TB, 0, stream>>>(srcI, dstI, ew, scal + 1, scal + 4,
                                              aS2, aD2, alpha2, m2, E, Etot, 1);
    edge_exp<<<blks(Etot), TB, 0, stream>>>(dstI, alpha2, m2, dn2, E, Etot, 1);
    edge_agg2<<<(unsigned)Etot, 64, 0, stream>>>(srcI, dstI, alpha2, dn2, h2,
                                                 out2, E);
    finalize<<<blks(N), TB, 0, stream>>>(out2, b2, (float*)d_out, N);
}